// SegRNN_43989055045730
// MI455X (gfx1250) — compile-verified
//
#include <hip/hip_runtime.h>
#include <math.h>

#define LW   15
#define KD   64
#define HD   16
#define NYC  15
#define NN   16384
#define SS   2048
#define NEGF (-1e30f)

// blocked log-semiring scan config: NN = CHUNKS * CLEN
#define CHUNKS 64
#define CLEN   256

typedef float v2f __attribute__((ext_vector_type(2)));
typedef float v8f __attribute__((ext_vector_type(8)));

static __device__ __forceinline__ v8f wmma4(v2f a, v2f b, v8f c) {
  // D(16x16,f32) += A(16x4,f32) * B(4x16,f32)
  // (neg_a, A, neg_b, B, c_mod, C, reuse_a, reuse_b)
  return __builtin_amdgcn_wmma_f32_16x16x4_f32(false, a, false, b, (short)0, c, false, false);
}

static __device__ __forceinline__ float fsig(float x) {
  return 1.0f / (1.0f + __expf(-x));
}
static __device__ __forceinline__ float ftanh(float x) {
  float e = __expf(-2.0f * fabsf(x));
  float r = (1.0f - e) / (1.0f + e);
  return copysignf(r, x);
}

// ---------------------------------------------------------------------------
// K1: G = data @ Wih^T + (bih + bhh), per direction.  One wave = one 16-row
// tile x all 64 gate columns (4 N-tiles), K=64 via 16 chained f32 WMMAs.
// ---------------------------------------------------------------------------
__global__ __launch_bounds__(256) void k_precompute_g(
    const float* __restrict__ data,
    const float* __restrict__ fWih, const float* __restrict__ fbih, const float* __restrict__ fbhh,
    const float* __restrict__ bWih, const float* __restrict__ bbih, const float* __restrict__ bbhh,
    float* __restrict__ Gf, float* __restrict__ Gb)
{
  const int lane = threadIdx.x & 31;
  const int wv   = threadIdx.x >> 5;
  const int wid  = blockIdx.x * 8 + wv;      // 0..2047
  const int dir  = wid >> 10;                // 0=fwd, 1=bwd
  const int tile = wid & 1023;               // 0..1023
  const float* Wih = dir ? bWih : fWih;
  const float* b0  = dir ? bbih : fbih;
  const float* b1  = dir ? bbhh : fbhh;
  float*       G   = dir ? Gb   : Gf;
  const int half = lane >> 4, ln = lane & 15;
  const int row  = tile * 16 + ln;           // A-matrix M = lane%16

  v8f acc[4];
#pragma unroll
  for (int jt = 0; jt < 4; ++jt) {
    float bias = b0[jt * 16 + ln] + b1[jt * 16 + ln];   // per-column (n = ln)
#pragma unroll
    for (int r = 0; r < 8; ++r) acc[jt][r] = bias;
  }
#pragma unroll
  for (int kb = 0; kb < 16; ++kb) {
    const int k = kb * 4 + 2 * half;         // A: K striped over vgpr + lane-half
    v2f a; a.x = data[row * 64 + k]; a.y = data[row * 64 + k + 1];
#pragma unroll
    for (int jt = 0; jt < 4; ++jt) {
      v2f b;                                  // B[k][n] = Wih[jt*16+ln][k]
      b.x = Wih[(jt * 16 + ln) * 64 + k];
      b.y = Wih[(jt * 16 + ln) * 64 + k + 1];
      acc[jt] = wmma4(a, b, acc[jt]);
    }
  }
#pragma unroll
  for (int jt = 0; jt < 4; ++jt)
#pragma unroll
    for (int r = 0; r < 8; ++r)
      G[(tile * 16 + r + 8 * half) * 64 + jt * 16 + ln] = acc[jt][r];
}

// ---------------------------------------------------------------------------
// K2: fused LSTMs + scoring head.  One wave owns 16 output positions
// [p0,p0+16): 32 fwd chains (14-row halo) + 16 bwd chains.  h lives in LDS
// (A-operand staging), c in registers (C/D layout).  Per step d:
//   gates = G[gather] + h @ Whh^T   (WMMA, N-tiles == i,f,g,o)
//   base  = fb @ V1^T + Z V3^T + Vb (WMMA, K=32)
//   scores/LSE over NY via lane-half butterflies -> A[p,d], T[p][y] (p==d<15)
// ---------------------------------------------------------------------------
#define WAVES2 4
__global__ __launch_bounds__(WAVES2 * 32) void k_lstm_scores(
    const float* __restrict__ Gf, const float* __restrict__ Gb,
    const float* __restrict__ fWhh, const float* __restrict__ bWhh,
    const float* __restrict__ fh0, const float* __restrict__ fc0,
    const float* __restrict__ bh0, const float* __restrict__ bc0,
    const float* __restrict__ Yenc, const float* __restrict__ Zenc,
    const float* __restrict__ VW,   const float* __restrict__ Vb,
    const float* __restrict__ WW,   const float* __restrict__ Wb,
    float* __restrict__ Aout, float* __restrict__ Ttab)
{
  __shared__ float ypart[NYC][16];
  __shared__ float hF[WAVES2][32][17];
  __shared__ float hB[WAVES2][16][17];

  const int tid  = threadIdx.x;
  const int lane = tid & 31, wv = tid >> 5;
  const int half = lane >> 4, ln = lane & 15;
  const int p0   = (blockIdx.x * WAVES2 + wv) * 16;

  // ypart[y][h'] = sum_t Y_enc[y,t] * V2[h',t]
  for (int idx = tid; idx < NYC * 16; idx += WAVES2 * 32) {
    int y = idx >> 4, hh = idx & 15;
    float s = 0.f;
#pragma unroll
    for (int t = 0; t < 32; ++t) s += Yenc[y * 32 + t] * VW[hh * 68 + 32 + t];
    ypart[y][hh] = s;
  }
#pragma unroll
  for (int n = 0; n < 16; ++n) {
    hF[wv][lane][n] = fh0[n];
    if (lane < 16) hB[wv][lane][n] = bh0[n];
  }
  __syncthreads();

  v8f cF[2], cB;
#pragma unroll
  for (int r = 0; r < 8; ++r) { cF[0][r] = fc0[ln]; cF[1][r] = fc0[ln]; cB[r] = bc0[ln]; }

  const float Wwn = WW[ln];
  const float Wbv = Wb[0];
  const float Vbn = Vb[ln];

  for (int d = 0; d < LW; ++d) {
    // ---------------- forward LSTM tiles (chains q = clamp(p0-14+row)) -----
#pragma unroll
    for (int ft = 0; ft < 2; ++ft) {
      v8f g[4];
      int xidx[8];
#pragma unroll
      for (int r = 0; r < 8; ++r) {
        int m = r + 8 * half;
        int q = p0 - 14 + ft * 16 + m;
        q = max(q, 0); q = min(q, NN - 1);
        xidx[r] = min(q + d, NN - 1);        // win_f clamp
      }
#pragma unroll
      for (int jt = 0; jt < 4; ++jt)
#pragma unroll
        for (int r = 0; r < 8; ++r)
          g[jt][r] = Gf[xidx[r] * 64 + jt * 16 + ln];
#pragma unroll
      for (int kb = 0; kb < 4; ++kb) {
        const int k = kb * 4 + 2 * half;
        v2f a; a.x = hF[wv][ft * 16 + ln][k]; a.y = hF[wv][ft * 16 + ln][k + 1];
#pragma unroll
        for (int jt = 0; jt < 4; ++jt) {
          v2f b;
          b.x = fWhh[(jt * 16 + ln) * 16 + k];
          b.y = fWhh[(jt * 16 + ln) * 16 + k + 1];
          g[jt] = wmma4(a, b, g[jt]);
        }
      }
#pragma unroll
      for (int r = 0; r < 8; ++r) {
        float iv = fsig(g[0][r]), fv = fsig(g[1][r]);
        float gv = ftanh(g[2][r]), ov = fsig(g[3][r]);
        float cv = fv * cF[ft][r] + iv * gv;
        cF[ft][r] = cv;
        hF[wv][ft * 16 + r + 8 * half][ln] = ov * ftanh(cv);
      }
    }
    // ---------------- backward LSTM tile (chains p = p0+row) --------------
    {
      v8f g[4];
      int xidx[8];
#pragma unroll
      for (int r = 0; r < 8; ++r) {
        int m = r + 8 * half;
        xidx[r] = max(p0 + m - d, 0);        // win_b clamp
      }
#pragma unroll
      for (int jt = 0; jt < 4; ++jt)
#pragma unroll
        for (int r = 0; r < 8; ++r)
          g[jt][r] = Gb[xidx[r] * 64 + jt * 16 + ln];
#pragma unroll
      for (int kb = 0; kb < 4; ++kb) {
        const int k = kb * 4 + 2 * half;
        v2f a; a.x = hB[wv][ln][k]; a.y = hB[wv][ln][k + 1];
#pragma unroll
        for (int jt = 0; jt < 4; ++jt) {
          v2f b;
          b.x = bWhh[(jt * 16 + ln) * 16 + k];
          b.y = bWhh[(jt * 16 + ln) * 16 + k + 1];
          g[jt] = wmma4(a, b, g[jt]);
        }
      }
#pragma unroll
      for (int r = 0; r < 8; ++r) {
        float iv = fsig(g[0][r]), fv = fsig(g[1][r]);
        float gv = ftanh(g[2][r]), ov = fsig(g[3][r]);
        float cv = fv * cB[r] + iv * gv;
        cB[r] = cv;
        hB[wv][r + 8 * half][ln] = ov * ftanh(cv);
      }
    }
    // ---------------- base = fb @ V1^T + Z V3^T + Vb (K=32, 8 WMMAs) ------
    v8f base;
    {
      float zb = Vbn;
#pragma unroll
      for (int j = 0; j < 4; ++j) zb += Zenc[d * 4 + j] * VW[ln * 68 + 64 + j];
#pragma unroll
      for (int r = 0; r < 8; ++r) base[r] = zb;
    }
    const int rf = max(p0 + ln - d, 0) - p0 + 14;   // fwd chain row for p=p0+ln, in [0,29]
#pragma unroll
    for (int kb = 0; kb < 8; ++kb) {
      const int k = kb * 4 + 2 * half;              // wave-uniform split at k=16
      v2f a;
      if (k < 16) { a.x = hF[wv][rf][k];      a.y = hF[wv][rf][k + 1]; }
      else        { a.x = hB[wv][ln][k - 16]; a.y = hB[wv][ln][k - 15]; }
      v2f b; b.x = VW[ln * 68 + k]; b.y = VW[ln * 68 + k + 1];
      base = wmma4(a, b, base);
    }
    // ---------------- scores, streaming LSE over NY ----------------------
    float mx[8], se[8];
#pragma unroll
    for (int r = 0; r < 8; ++r) { mx[r] = NEGF; se[r] = 0.f; }
    for (int y = 0; y < NYC; ++y) {
      const float yp = ypart[y][ln];
#pragma unroll
      for (int r = 0; r < 8; ++r) {
        float v = ftanh(base[r] + yp) * Wwn;
        v += __shfl_xor(v, 1, 32);
        v += __shfl_xor(v, 2, 32);
        v += __shfl_xor(v, 4, 32);
        v += __shfl_xor(v, 8, 32);              // reduce over h' (16-lane half)
        const float s = v + Wbv;
        const float m2 = fmaxf(mx[r], s);
        se[r] = se[r] * __expf(mx[r] - m2) + __expf(s - m2);
        mx[r] = m2;
        const int pg = p0 + r + 8 * half;
        if (pg == d && pg < LW && ln == 0) Ttab[pg * NYC + y] = s;  // gold table
      }
    }
#pragma unroll
    for (int r = 0; r < 8; ++r) {
      const int pg = p0 + r + 8 * half;
      const float Aval = (pg - d >= 0) ? (mx[r] + __logf(se[r])) : NEGF;
      if (ln == 0) Aout[pg * LW + d] = Aval;
    }
  }
}

// ---------------------------------------------------------------------------
// K3a: blocked scan, phase 1.  DP step p is a log-semiring operator
// (row0 = A[p,:], rows i>=1 = shifted identity).  Chunk matrix column j =
// recurrence run from basis vector e_j.  One THREAD per (chunk, column):
// 960 threads, h[15]+v[15] in registers, zero cross-lane traffic.
// ---------------------------------------------------------------------------
__global__ __launch_bounds__(64) void k_scan_chunks(
    const float* __restrict__ Aout, float* __restrict__ Mc)
{
  const int t = blockIdx.x * 64 + threadIdx.x;   // 0..959 (grid = 15 blocks)
  const int c = t / NYC;                         // chunk 0..63
  const int j = t % NYC;                         // column 0..14

  float h[LW];
#pragma unroll
  for (int i = 0; i < LW; ++i) h[i] = (i == j) ? 0.f : NEGF;

  const int pbeg = c * CLEN;
  for (int p = pbeg; p < pbeg + CLEN; ++p) {
    const float* Ar = Aout + p * LW;
    float v[LW];
    float m = NEGF;
#pragma unroll
    for (int k = 0; k < LW; ++k) {
      v[k] = h[k] + Ar[k];
      m = fmaxf(m, v[k]);
    }
    float s = 0.f;
#pragma unroll
    for (int k = 0; k < LW; ++k) s += __expf(v[k] - m);
    const float nv = m + __logf(s);
#pragma unroll
    for (int i = LW - 1; i >= 1; --i) h[i] = h[i - 1];  // shift-in
    h[0] = nv;
  }
#pragma unroll
  for (int i = 0; i < LW; ++i) Mc[(c * LW + i) * LW + j] = h[i];
}

// ---------------------------------------------------------------------------
// K3b: phase 2 — one wave applies the 64 chunk matrices to hist0 = [0,NEG..]
// (lane i owns hist[i]; h[j] broadcast by shuffle), then the gold-segment
// reduction and the final scalar  out = logZ - indiv.
// ---------------------------------------------------------------------------
__global__ void k_scan_combine(const float* __restrict__ Mc, const float* __restrict__ Ttab,
                               const int* __restrict__ tags, const int* __restrict__ lengths,
                               float* __restrict__ out)
{
  const int lane = threadIdx.x;                 // 32 threads
  const int i    = min(lane, NYC - 1);          // clamp so lanes 15..31 load safely
  float h = (lane == 0) ? 0.f : NEGF;

  for (int c = 0; c < CHUNKS; ++c) {
    float v[LW];
    float m = NEGF;
#pragma unroll
    for (int j = 0; j < LW; ++j) {
      const float hj = __shfl(h, j, 32);        // hist[j] from lane j
      v[j] = Mc[(c * LW + i) * LW + j] + hj;
      m = fmaxf(m, v[j]);
    }
    float s = 0.f;
#pragma unroll
    for (int j = 0; j < LW; ++j) s += __expf(v[j] - m);
    h = m + __logf(s);                          // lanes >=15 compute clamped junk, never read
  }
  const float logZ = __shfl(h, 0, 32);

  float acc = 0.f;
  for (int s = lane; s < SS; s += 32) {
    const int len = lengths[s];
    if (len < LW) acc += Ttab[(len - 1) * NYC + tags[s]];
  }
  acc += __shfl_xor(acc, 1, 32);
  acc += __shfl_xor(acc, 2, 32);
  acc += __shfl_xor(acc, 4, 32);
  acc += __shfl_xor(acc, 8, 32);
  acc += __shfl_xor(acc, 16, 32);
  if (lane == 0) out[0] = logZ - acc;
}

// ---------------------------------------------------------------------------
extern "C" void kernel_launch(void* const* d_in, const int* in_sizes, int n_in,
                              void* d_out, int out_size, void* d_ws, size_t ws_size,
                              hipStream_t stream) {
  const float* data = (const float*)d_in[0];
  const float* fWih = (const float*)d_in[1];
  const float* fWhh = (const float*)d_in[2];
  const float* fbih = (const float*)d_in[3];
  const float* fbhh = (const float*)d_in[4];
  const float* bWih = (const float*)d_in[5];
  const float* bWhh = (const float*)d_in[6];
  const float* bbih = (const float*)d_in[7];
  const float* bbhh = (const float*)d_in[8];
  const float* fh0  = (const float*)d_in[9];
  const float* fc0  = (const float*)d_in[10];
  const float* bh0  = (const float*)d_in[11];
  const float* bc0  = (const float*)d_in[12];
  const float* Yenc = (const float*)d_in[13];
  const float* Zenc = (const float*)d_in[14];
  const float* VW   = (const float*)d_in[15];
  const float* Vb   = (const float*)d_in[16];
  const float* WW   = (const float*)d_in[17];
  const float* Wb   = (const float*)d_in[18];
  const int*   tags    = (const int*)d_in[19];
  const int*   lengths = (const int*)d_in[20];

  char* ws = (char*)d_ws;
  float* Gf   = (float*)(ws);                       // N*64 f32 = 4 MB
  float* Gb   = (float*)(ws + 4194304);             // 4 MB
  float* Aout = (float*)(ws + 8388608);             // N*15 f32 = 0.94 MB
  float* Ttab = (float*)(ws + 8388608 + 983040);    // 15*15 f32 (pad to 4 KB)
  float* Mc   = (float*)(ws + 8388608 + 983040 + 4096);  // 64*15*15 f32 = 57.6 KB

  k_precompute_g<<<256, 256, 0, stream>>>(data, fWih, fbih, fbhh, bWih, bbih, bbhh, Gf, Gb);
  k_lstm_scores<<<256, WAVES2 * 32, 0, stream>>>(Gf, Gb, fWhh, bWhh, fh0, fc0, bh0, bc0,
                                                 Yenc, Zenc, VW, Vb, WW, Wb, Aout, Ttab);
  k_scan_chunks<<<15, 64, 0, stream>>>(Aout, Mc);
  k_scan_combine<<<1, 32, 0, stream>>>(Mc, Ttab, tags, lengths, (float*)d_out);
}